// ContinuousNormalizingFlow_81930796139202
// MI455X (gfx1250) — compile-verified
//
#include <hip/hip_runtime.h>

#define BATCH 4096
#define DSZ 16
#define WID 128
#define NSTEPS 10

typedef float v2f __attribute__((ext_vector_type(2)));
typedef float v8f __attribute__((ext_vector_type(8)));

// workspace / LDS float offsets (weights region is copied contiguously to LDS)
#define OFF_B1   0        // 4kc*8nt*64  = 2048   (W1[:, :16] as B-fragments)
#define OFF_B2   2048     // 32*8*64     = 16384  (W2 as B-fragments)
#define OFF_B3   18432    // 32*1*64     = 2048   (W3 as B-fragments)
#define OFF_b1   20480    // 128
#define OFF_b2   20608    // 128
#define OFF_b3   20736    // 16
#define OFF_w1t  20752    // 128  (W1[:,16], the t column)
#define LDS_WEIGHTS 20880
#define OFF_B4   20880    // 32*8*64 = 16384 (trace matrix C, stays in global/L2)
#define WS_FLOATS 37264

#define SSTRIDE 132                 // padded row stride for 16x128 scratch tiles
#define WAVE_SCRATCH (3*16*SSTRIDE + 16)   // h1s, d1s, h2s, trb = 6352 floats

__device__ __forceinline__ v8f wmma4(v2f a, v2f b, v8f c) {
  // D = A(16x4 f32) * B(4x16 f32) + C(16x16 f32)
  return __builtin_amdgcn_wmma_f32_16x16x4_f32(false, a, false, b, (short)0, c,
                                               false, false);
}

__device__ __forceinline__ v8f vzero8() {
  v8f z = {0.f, 0.f, 0.f, 0.f, 0.f, 0.f, 0.f, 0.f};
  return z;
}

// sigma(z) with v_exp_f32 + v_rcp_f32 (avoids the full IEEE divide expansion)
__device__ __forceinline__ float fast_sigmoid(float z) {
  return __builtin_amdgcn_rcpf(1.f + __expf(-z));
}

// ---------------------------------------------------------------------------
// Pack kernel: build B-matrix WMMA fragments + trace matrix C in workspace.
// Fragment layout (per 16x16 output tile, K-chunk of 4):
//   frag[(kc*NT + nt)*64 + lane*2 + j] = B[k][n],
//   k = kc*4 + (lane>=16 ? 2 : 0) + j,  n = nt*16 + (lane & 15)
// (mirrors the documented 32-bit A-matrix 16x4 layout, lanes <-> columns)
// ---------------------------------------------------------------------------
__global__ void cnf_pack(const float* __restrict__ W1, const float* __restrict__ W2,
                         const float* __restrict__ W3, const float* __restrict__ b1,
                         const float* __restrict__ b2, const float* __restrict__ b3,
                         float* __restrict__ ws) {
  int idx = blockIdx.x * blockDim.x + threadIdx.x;
  if (idx >= WS_FLOATS) return;
  float val;
  if (idx < OFF_B2) {                       // B1: z1 = y @ W1y^T  -> B[k][n] = W1[n,k]
    int i = idx - OFF_B1, fr = i >> 6, w = i & 63, lane = w >> 1, j = w & 1;
    int kc = fr >> 3, nt = fr & 7;
    int k = kc * 4 + (lane >> 4) * 2 + j, n = nt * 16 + (lane & 15);
    val = W1[n * 17 + k];
  } else if (idx < OFF_B3) {                // B2: z2 = h1 @ W2^T -> B[k][n] = W2[n,k]
    int i = idx - OFF_B2, fr = i >> 6, w = i & 63, lane = w >> 1, j = w & 1;
    int kc = fr >> 3, nt = fr & 7;
    int k = kc * 4 + (lane >> 4) * 2 + j, n = nt * 16 + (lane & 15);
    val = W2[n * 128 + k];
  } else if (idx < OFF_b1) {                // B3: f = h2 @ W3^T -> B[k][n] = W3[n,k]
    int i = idx - OFF_B3, kc = i >> 6, w = i & 63, lane = w >> 1, j = w & 1;
    int k = kc * 4 + (lane >> 4) * 2 + j, n = lane & 15;
    val = W3[n * 128 + k];
  } else if (idx < OFF_b2) {
    val = b1[idx - OFF_b1];
  } else if (idx < OFF_b3) {
    val = b2[idx - OFF_b2];
  } else if (idx < OFF_w1t) {
    val = b3[idx - OFF_b3];
  } else if (idx < OFF_B4) {
    val = W1[(idx - OFF_w1t) * 17 + 16];    // t-column of W1
  } else {                                  // B4: U = d1 @ C^T -> B[b][a] = C[a,b]
    int i = idx - OFF_B4, fr = i >> 6, w = i & 63, lane = w >> 1, j = w & 1;
    int kc = fr >> 3, nt = fr & 7;
    int b = kc * 4 + (lane >> 4) * 2 + j;   // k index (over d1)
    int a = nt * 16 + (lane & 15);          // n index (over d2)
    // C[a,b] = M[b,a] * W2[a,b],  M = W1y @ W3  (128x128)
    float m = 0.f;
#pragma unroll
    for (int ii = 0; ii < 16; ++ii) m += W1[b * 17 + ii] * W3[ii * 128 + a];
    val = m * W2[a * 128 + b];
  }
  ws[idx] = val;
}

// ---------------------------------------------------------------------------
// One vector-field evaluation: (f, tr(df/dy)) for a 16-sample tile per wave.
// yA / fA are 16x16 tiles in f32 WMMA A-format (4 K-chunks of v2f per lane).
// ---------------------------------------------------------------------------
__device__ __forceinline__ void vf_eval(
    const float* sB1, const float* sB2, const float* sB3, const float* sb1,
    const float* sb2, const float* sb3, const float* sw1t,
    const float* __restrict__ gB4, float* h1s, float* d1s, float* h2s,
    float* trb, int lane, float t, const v2f yA[4], v2f fA[4], float& fl) {
  const int nloc = lane & 15;
  const int hl = lane >> 4;

  // ---- layer 1: z1 = [y,t] @ W1^T + b1 ; h1 = silu(z1), d1 = silu'(z1) ----
#pragma unroll
  for (int nt = 0; nt < 8; ++nt) {
    v8f acc = vzero8();
#pragma unroll
    for (int kc = 0; kc < 4; ++kc) {
      v2f b = *(const v2f*)(sB1 + (kc * 8 + nt) * 64 + lane * 2);
      acc = wmma4(yA[kc], b, acc);
    }
    float bias = sb1[nt * 16 + nloc] + t * sw1t[nt * 16 + nloc];
#pragma unroll
    for (int r = 0; r < 8; ++r) {
      float z = acc[r] + bias;
      float s = fast_sigmoid(z);
      int row = r + 8 * hl;                       // D-format: m = r + 8*half
      h1s[row * SSTRIDE + nt * 16 + nloc] = z * s;
      d1s[row * SSTRIDE + nt * 16 + nloc] = s * (1.f + z * (1.f - s));
    }
  }
  __syncthreads();

  // ---- layer 2 (z2) fused with trace GEMM (U = d1 @ C^T) ----
  v8f tracc = vzero8();
#pragma unroll 2
  for (int nt = 0; nt < 8; ++nt) {
    v8f accz = vzero8();
    v8f accu = vzero8();
    for (int kc = 0; kc < 32; ++kc) {
      v2f a = *(const v2f*)(h1s + nloc * SSTRIDE + kc * 4 + hl * 2);
      v2f b = *(const v2f*)(sB2 + (kc * 8 + nt) * 64 + lane * 2);
      accz = wmma4(a, b, accz);
      v2f ad = *(const v2f*)(d1s + nloc * SSTRIDE + kc * 4 + hl * 2);
      v2f b4 = *(const v2f*)(gB4 + (kc * 8 + nt) * 64 + lane * 2);
      accu = wmma4(ad, b4, accu);
    }
    float bias = sb2[nt * 16 + nloc];
#pragma unroll
    for (int r = 0; r < 8; ++r) {
      float z = accz[r] + bias;
      float s = fast_sigmoid(z);
      h2s[(r + 8 * hl) * SSTRIDE + nt * 16 + nloc] = z * s;
      tracc[r] += (s * (1.f + z * (1.f - s))) * accu[r];  // d2 ⊙ U
    }
  }
  __syncthreads();

  // ---- trace: reduce over the 16 lanes of each wave half ----
#pragma unroll
  for (int r = 0; r < 8; ++r) {
    float v = tracc[r];
    v += __shfl_xor(v, 1, 32);
    v += __shfl_xor(v, 2, 32);
    v += __shfl_xor(v, 4, 32);
    v += __shfl_xor(v, 8, 32);
    tracc[r] = v;
  }
  if (nloc == 0) {                       // lanes 0 (m=0..7) and 16 (m=8..15)
#pragma unroll
    for (int r = 0; r < 8; ++r) trb[hl * 8 + r] = tracc[r];
  }

  // ---- layer 3: f = h2 @ W3^T + b3 ----
  v8f accf = vzero8();
  for (int kc = 0; kc < 32; ++kc) {
    v2f a = *(const v2f*)(h2s + nloc * SSTRIDE + kc * 4 + hl * 2);
    v2f b = *(const v2f*)(sB3 + kc * 64 + lane * 2);
    accf = wmma4(a, b, accf);
  }
  __syncthreads();
  // bounce f D-format -> A-format through h1s region
#pragma unroll
  for (int r = 0; r < 8; ++r)
    h1s[(r + 8 * hl) * SSTRIDE + nloc] = accf[r] + sb3[nloc];
  __syncthreads();
#pragma unroll
  for (int kc = 0; kc < 4; ++kc)
    fA[kc] = *(const v2f*)(h1s + nloc * SSTRIDE + kc * 4 + hl * 2);
  fl = trb[nloc];
}

// ---------------------------------------------------------------------------
// Main kernel: 32 blocks x 256 threads (8 wave32). Each wave integrates a
// 16-sample tile through all 10 ReversibleHeun steps with state in registers.
// ---------------------------------------------------------------------------
extern __shared__ float smem[];

__global__ void __launch_bounds__(256) cnf_main(const float* __restrict__ y_in,
                                               const float* __restrict__ ws,
                                               float* __restrict__ out) {
  const int tid = threadIdx.x;
  const int lane = tid & 31;
  const int wave = tid >> 5;

  for (int i = tid; i < LDS_WEIGHTS; i += 256) smem[i] = ws[i];
  __syncthreads();

  const float* sB1 = smem + OFF_B1;
  const float* sB2 = smem + OFF_B2;
  const float* sB3 = smem + OFF_B3;
  const float* sb1 = smem + OFF_b1;
  const float* sb2 = smem + OFF_b2;
  const float* sb3 = smem + OFF_b3;
  const float* sw1t = smem + OFF_w1t;
  const float* gB4 = ws + OFF_B4;
  float* wbase = smem + LDS_WEIGHTS + wave * WAVE_SCRATCH;
  float* h1s = wbase;
  float* d1s = wbase + 16 * SSTRIDE;
  float* h2s = wbase + 32 * SSTRIDE;
  float* trb = wbase + 48 * SSTRIDE;

  const int nloc = lane & 15;
  const int hl = lane >> 4;
  const int row = blockIdx.x * 128 + wave * 16 + nloc;

  v2f yc[4], yh[4], fy[4], f1[4];
#pragma unroll
  for (int kc = 0; kc < 4; ++kc) {
    yc[kc] = *(const v2f*)(y_in + row * 16 + kc * 4 + hl * 2);
    yh[kc] = yc[kc];
  }
  float lc = 0.f, lh = 0.f, fl = 0.f, fl1 = 0.f;

  vf_eval(sB1, sB2, sB3, sb1, sb2, sb3, sw1t, gB4, h1s, d1s, h2s, trb, lane,
          0.0f, yc, fy, fl);

  const float dt = 0.1f;
  for (int i = 0; i < NSTEPS; ++i) {
    float t1 = (float)(i + 1) * dt;
#pragma unroll
    for (int kc = 0; kc < 4; ++kc) yh[kc] = 2.f * yc[kc] - yh[kc] + dt * fy[kc];
    lh = 2.f * lc - lh + dt * fl;
    vf_eval(sB1, sB2, sB3, sb1, sb2, sb3, sw1t, gB4, h1s, d1s, h2s, trb, lane,
            t1, yh, f1, fl1);
#pragma unroll
    for (int kc = 0; kc < 4; ++kc)
      yc[kc] = yc[kc] + (0.5f * dt) * (fy[kc] + f1[kc]);
    lc = lc + 0.5f * dt * (fl + fl1);
#pragma unroll
    for (int kc = 0; kc < 4; ++kc) fy[kc] = f1[kc];
    fl = fl1;
  }

#pragma unroll
  for (int kc = 0; kc < 4; ++kc)
    *(v2f*)(out + row * 16 + kc * 4 + hl * 2) = yc[kc];
  if (hl == 0) out[BATCH * DSZ + blockIdx.x * 128 + wave * 16 + nloc] = lc;
}

extern "C" void kernel_launch(void* const* d_in, const int* in_sizes, int n_in,
                              void* d_out, int out_size, void* d_ws,
                              size_t ws_size, hipStream_t stream) {
  (void)in_sizes; (void)n_in; (void)out_size; (void)ws_size;
  const float* y  = (const float*)d_in[0];
  const float* W1 = (const float*)d_in[1];
  const float* b1 = (const float*)d_in[2];
  const float* W2 = (const float*)d_in[3];
  const float* b2 = (const float*)d_in[4];
  const float* W3 = (const float*)d_in[5];
  const float* b3 = (const float*)d_in[6];
  float* ws = (float*)d_ws;
  float* out = (float*)d_out;

  cnf_pack<<<(WS_FLOATS + 255) / 256, 256, 0, stream>>>(W1, W2, W3, b1, b2, b3, ws);

  size_t smembytes = (size_t)(LDS_WEIGHTS + 8 * WAVE_SCRATCH) * sizeof(float);
  (void)hipFuncSetAttribute((const void*)cnf_main,
                            hipFuncAttributeMaxDynamicSharedMemorySize,
                            (int)smembytes);
  cnf_main<<<BATCH / 128, 256, smembytes, stream>>>(y, ws, out);
}